// DiceLoss_16200616640590
// MI455X (gfx1250) — compile-verified
//
#include <hip/hip_runtime.h>
#include <math.h>

// Problem constants (from reference)
#define CCH 16
#define WT  593          // (C+2)*C + C + C*C + C + C + 1
#define HH  128
#define WWD 128
#define HW  (HH*WWD)
#define NB  4
#define NO  32

typedef __attribute__((ext_vector_type(2))) float v2f;
typedef __attribute__((ext_vector_type(4))) float v4f;
typedef __attribute__((ext_vector_type(8))) float v8f;

__device__ __forceinline__ float sigmoidf_(float x) { return 1.f / (1.f + __expf(-x)); }

// One workgroup per (b,o) instance. 256 threads = 8 wave32.
// Per 16-pixel tile: layer1 = 5x V_WMMA_F32_16X16X4_F32 (K=18 padded to 20),
// layer2 = 4x (h1 re-laid out via per-wave LDS staging), layer3 = VALU dot +
// half-wave shuffle. All fp32, matching the reference numerics.
__global__ __launch_bounds__(256) void condinst_dice_kernel(
    const float* __restrict__ seg_feat,     // (B, C, H, W)
    const float* __restrict__ conv_weight,  // (B, WT, H, W)
    const int*   __restrict__ mask,         // (B, O)
    const int*   __restrict__ ind,          // (B, O)
    const float* __restrict__ target,       // (B, O, H, W)
    float*       __restrict__ acc)          // acc[b*3+{0,1,2}] = inter, sum p^2, sum t^2
{
    const int bo = blockIdx.x;
    const int b  = bo >> 5;

    if (mask[bo] == 0) return;  // masked instance contributes exactly zero

    __shared__ float w1p[16][20];   // W1 padded K 18->20
    __shared__ float b1s[16];
    __shared__ float w2s[16][16];
    __shared__ float b2s[16];
    __shared__ float w3s[16];
    __shared__ float b3s;
    __shared__ float hst[8][16 * 20];  // per-wave h1 staging, [pixel][channel], stride 20

    const int tid = threadIdx.x;
    const int pix = ind[bo];
    const float xi = (float)(pix & 127);
    const float yi = (float)(pix >> 7);

    // Gather the 593 instance weights (stride HW between "channels")
    const float* cw = conv_weight + (size_t)b * WT * HW + pix;
    for (int j = tid; j < WT; j += 256) {
        float v = cw[(size_t)j * HW];
        if (j < 288)        w1p[j / 18][j % 18] = v;
        else if (j < 304)   b1s[j - 288] = v;
        else if (j < 560)   w2s[(j - 304) >> 4][(j - 304) & 15] = v;
        else if (j < 576)   b2s[j - 560] = v;
        else if (j < 592)   w3s[j - 576] = v;
        else                b3s = v;
    }
    if (tid < 32) w1p[tid >> 1][18 + (tid & 1)] = 0.f;  // zero the K padding
    __syncthreads();

    const int lane = tid & 31;
    const int wv   = tid >> 5;       // wave id 0..7
    const int m    = lane & 15;      // A row / B-D column index
    const int hi   = lane >> 4;      // half-wave select
    const int kh   = hi * 2;         // K sub-offset inside a 4-chunk

    // A-matrix register images per ISA 16x4 f32 layout:
    //   lanes 0-15: VGPR0=K0, VGPR1=K1 ; lanes 16-31: VGPR0=K2, VGPR1=K3
    v2f A1[5], A2[4];
#pragma unroll
    for (int kc = 0; kc < 5; ++kc) { A1[kc].x = w1p[m][4*kc+kh]; A1[kc].y = w1p[m][4*kc+kh+1]; }
#pragma unroll
    for (int kc = 0; kc < 4; ++kc) { A2[kc].x = w2s[m][4*kc+kh]; A2[kc].y = w2s[m][4*kc+kh+1]; }

    float bias1[8], bias2[8], w3r[8];
#pragma unroll
    for (int r = 0; r < 8; ++r) {
        bias1[r] = b1s[r + 8*hi];
        bias2[r] = b2s[r + 8*hi];
        w3r[r]   = w3s[r + 8*hi];
    }
    const float b3v = b3s;

    // Per-lane LDS staging addresses (dword indices), hoisted out of the loop.
    float* hs = hst[wv];
    float* st_ptr = hs + m * 20 + 8 * hi;   // contiguous 8 floats: rows k=8hi..8hi+7 of pixel m
    const float* ld_ptr = hs + m * 20 + kh; // + 4*kc gives the (k, k+1) pair this lane needs

    const float* seg = seg_feat + (size_t)b * CCH * HW;
    const float* tgt = target + (size_t)bo * HW;

    float s_it = 0.f, s_pp = 0.f, s_tt = 0.f;

    // 1024 tiles of 16 pixels; 8 waves round-robin (uniform -> EXEC all-1 at WMMA)
    for (int t = wv; t < HW / 16; t += 8) {
        const int p = t * 16 + m;                       // this lane's pixel (= column N)
        const float xr = ((float)(p & 127) - xi) * (1.f / 128.f);
        const float yr = ((float)(p >> 7)  - yi) * (1.f / 128.f);

        // ---- layer 1: h1_pre = W1(16x20) x F(20x16)
        v8f acc1 = {};
#pragma unroll
        for (int kc = 0; kc < 4; ++kc) {
            v2f bb;
            bb.x = seg[(size_t)(4*kc + kh) * HW + p];
            bb.y = seg[(size_t)(4*kc + kh + 1) * HW + p];
            acc1 = __builtin_amdgcn_wmma_f32_16x16x4_f32(false, A1[kc], false, bb,
                                                         (short)0, acc1, false, false);
        }
        {   // channels 16,17 = x_rel,y_rel ; 18,19 = zero pad (high lane-half)
            v2f bb; bb.x = hi ? 0.f : xr; bb.y = hi ? 0.f : yr;
            acc1 = __builtin_amdgcn_wmma_f32_16x16x4_f32(false, A1[4], false, bb,
                                                         (short)0, acc1, false, false);
        }
        float h1[8];
#pragma unroll
        for (int r = 0; r < 8; ++r) h1[r] = fmaxf(acc1[r] + bias1[r], 0.f);

        // ---- D-layout -> B-layout via per-wave LDS staging in [pixel][channel]
        // order: this lane's 8 values are contiguous (2x ds_store_b128), and each
        // B chunk it needs back is a contiguous pair (4x ds_load_b64). DS ops from
        // one wave execute in order; wave_barrier() fences compiler scheduling.
        v4f s0; s0.x = h1[0]; s0.y = h1[1]; s0.z = h1[2]; s0.w = h1[3];
        v4f s1; s1.x = h1[4]; s1.y = h1[5]; s1.z = h1[6]; s1.w = h1[7];
        *(v4f*)(st_ptr)     = s0;
        *(v4f*)(st_ptr + 4) = s1;
        __builtin_amdgcn_wave_barrier();

        v8f acc2 = {};
#pragma unroll
        for (int kc = 0; kc < 4; ++kc) {
            v2f bb = *(const v2f*)(ld_ptr + 4 * kc);
            acc2 = __builtin_amdgcn_wmma_f32_16x16x4_f32(false, A2[kc], false, bb,
                                                         (short)0, acc2, false, false);
        }
        __builtin_amdgcn_wave_barrier();  // reads done before next tile's stores

        // ---- layer 3 dot + sigmoid + dice partials
        float part = 0.f;
#pragma unroll
        for (int r = 0; r < 8; ++r) part += w3r[r] * fmaxf(acc2[r] + bias2[r], 0.f);
        part += __shfl_xor(part, 16, 32);               // combine the two M-halves
        if (!hi) {                                      // one lane per pixel accumulates
            const float pr = sigmoidf_(part + b3v);
            const float tg = tgt[p];
            s_it += pr * tg;
            s_pp += pr * pr;
            s_tt += tg * tg;
        }
    }

    // reduce lanes 0..15 (high lanes hold zeros), then atomically fold per batch
#pragma unroll
    for (int d = 8; d >= 1; d >>= 1) {
        s_it += __shfl_xor(s_it, d, 32);
        s_pp += __shfl_xor(s_pp, d, 32);
        s_tt += __shfl_xor(s_tt, d, 32);
    }
    if (lane == 0) {
        atomicAdd(&acc[b * 3 + 0], s_it);
        atomicAdd(&acc[b * 3 + 1], s_pp);
        atomicAdd(&acc[b * 3 + 2], s_tt);
    }
}

__global__ void dice_finalize_kernel(const float* __restrict__ acc, float* __restrict__ out)
{
    if (threadIdx.x == 0 && blockIdx.x == 0) {
        float s = 0.f;
        for (int b = 0; b < NB; ++b) {
            float it = acc[b * 3 + 0];
            float pp = acc[b * 3 + 1];
            float tt = acc[b * 3 + 2];
            s += 1.f - (2.f * it + 1.f) / (pp + tt + 1.f);
        }
        *out = s * (1.f / (float)NB);
    }
}

extern "C" void kernel_launch(void* const* d_in, const int* in_sizes, int n_in,
                              void* d_out, int out_size, void* d_ws, size_t ws_size,
                              hipStream_t stream) {
    (void)in_sizes; (void)n_in; (void)out_size; (void)ws_size;
    const float* seg  = (const float*)d_in[0];
    const float* cwm  = (const float*)d_in[1];
    const int*   mask = (const int*)d_in[2];
    const int*   ind  = (const int*)d_in[3];
    const float* tgt  = (const float*)d_in[4];
    float* acc = (float*)d_ws;

    hipMemsetAsync(acc, 0, NB * 3 * sizeof(float), stream);
    condinst_dice_kernel<<<NB * NO, 256, 0, stream>>>(seg, cwm, mask, ind, tgt, acc);
    dice_finalize_kernel<<<1, 1, 0, stream>>>(acc, (float*)d_out);
}